// HSTUBlockPreprocessor_17918603559567
// MI455X (gfx1250) — compile-verified
//
#include <hip/hip_runtime.h>
#include <hip/hip_bf16.h>
#include <stdint.h>

#define DD 256  // feature dim (row = 1 KB fp32)

// 16-byte payload type the async b128 builtins expect: int ext_vector(4),
// global side in AS1, LDS side in AS3.
typedef int v4i __attribute__((ext_vector_type(4)));
typedef __attribute__((address_space(1))) v4i as1_v4i;
typedef __attribute__((address_space(3))) v4i as3_v4i;

// ---- CDNA5 async global<->LDS copy primitives (ASYNCcnt path) --------------
// cpol = 1 -> TH_NT (non-temporal): 537 MB streamed once, don't pollute L2.

__device__ __forceinline__ void async_load_b128(const float* g, float* lds) {
#if __has_builtin(__builtin_amdgcn_global_load_async_to_lds_b128)
  __builtin_amdgcn_global_load_async_to_lds_b128((as1_v4i*)g, (as3_v4i*)lds, 0, 1);
#else
  asm volatile("global_load_async_to_lds_b128 %0, %1, off th:TH_LOAD_NT"
               :
               : "v"((as3_v4i*)lds), "v"(g)
               : "memory");
#endif
}

__device__ __forceinline__ void async_store_b128(float* g, const float* lds) {
#if __has_builtin(__builtin_amdgcn_global_store_async_from_lds_b128)
  __builtin_amdgcn_global_store_async_from_lds_b128((as1_v4i*)g, (as3_v4i*)lds, 0, 1);
#else
  asm volatile("global_store_async_from_lds_b128 %0, %1, off th:TH_STORE_NT"
               :
               : "v"(g), "v"((as3_v4i*)lds)
               : "memory");
#endif
}

__device__ __forceinline__ void wait_asynccnt0() {
#if __has_builtin(__builtin_amdgcn_s_wait_asynccnt)
  __builtin_amdgcn_s_wait_asynccnt(0);
#else
  asm volatile("s_wait_asynccnt 0" ::: "memory");
#endif
}

// ---- Kernel: bulk interleaved copy via async DMA through LDS ---------------
// (first in file so the disasm snippet shows the async-tensor path)
// 8 waves/block; each wave owns 4 output rows (4 KB LDS slice).
// Global seq row r (token t=r>>1, item/action by r&1, sample s) maps to
// output row r + 2*s + 2 — fully coalesced 1 KB row moves.
__global__ __launch_bounds__(256) void seq_copy_kernel(
    const float* __restrict__ item, const float* __restrict__ action,
    const int* __restrict__ off, float* __restrict__ out,
    int Bn, int rows) {
  __shared__ float stage[8 * 4 * DD];  // 32 KB staging (per-wave private slices)
  __shared__ int sh_off[1025];

  for (int i = threadIdx.x; i <= Bn; i += blockDim.x) sh_off[i] = off[i];
  __syncthreads();

  const int wave = threadIdx.x >> 5;
  const int lane = threadIdx.x & 31;
  const int baseRow = blockIdx.x * 32 + wave * 4;
  float* wstage = &stage[wave * 4 * DD];

  float* dsts[4];
  int valid[4];

#pragma unroll
  for (int j = 0; j < 4; ++j) {
    const int r = baseRow + j;
    valid[j] = (r < rows);
    dsts[j] = nullptr;
    if (!valid[j]) continue;
    const int t = r >> 1;
    // largest s with sh_off[s] <= t (7 steps for B=128)
    int lo = 0, hi = Bn - 1;
    while (lo < hi) {
      const int mid = (lo + hi + 1) >> 1;
      if (sh_off[mid] <= t) lo = mid; else hi = mid - 1;
    }
    const float* src = ((r & 1) ? action : item) + (size_t)t * DD + lane * 4;
    float* lbuf = wstage + j * DD + lane * 4;
    // 1 KB row = 32 lanes x 2 x B128
    async_load_b128(src, lbuf);
    async_load_b128(src + 128, lbuf + 128);
    dsts[j] = out + (size_t)(r + 2 * lo + 2) * DD + lane * 4;
  }

  wait_asynccnt0();  // loads have landed in this wave's LDS slice

#pragma unroll
  for (int j = 0; j < 4; ++j) {
    if (!valid[j]) continue;
    float* lbuf = wstage + j * DD + lane * 4;
    async_store_b128(dsts[j], lbuf);
    async_store_b128(dsts[j] + 128, lbuf + 128);
  }
  // outstanding async stores drain at s_endpgm's implicit wait-idle
}

// ---- Kernel: offsets scan + out_len ----------------------------------------
// B is tiny (128); a serial scan by one lane costs ~nothing once per launch.
__global__ void scan_lengths_kernel(const int* __restrict__ lens, int Bn,
                                    int* __restrict__ off,
                                    int* __restrict__ out_len) {
  if (threadIdx.x == 0) {
    int acc = 0;
    for (int s = 0; s < Bn; ++s) { off[s] = acc; acc += lens[s]; }
    off[Bn] = acc;
  }
  for (int s = threadIdx.x; s < Bn; s += blockDim.x) {
    out_len[s] = 2 * lens[s] + 2;
  }
}

// ---- Kernel: context rows (2 rows per sample, 256 KB total) ----------------
__global__ __launch_bounds__(DD) void ctx_copy_kernel(
    const float* __restrict__ c0, const float* __restrict__ c1,
    const int* __restrict__ off, float* __restrict__ out) {
  const int s = blockIdx.x;
  const int i = threadIdx.x;  // 0..255 == DD
  const size_t row0 = (size_t)2 * (size_t)off[s] + (size_t)2 * s;
  const float v0 = __builtin_nontemporal_load(&c0[(size_t)s * DD + i]);
  const float v1 = __builtin_nontemporal_load(&c1[(size_t)s * DD + i]);
  __builtin_nontemporal_store(v0, &out[row0 * DD + i]);
  __builtin_nontemporal_store(v1, &out[(row0 + 1) * DD + i]);
}

// ---- Host launch -----------------------------------------------------------
extern "C" void kernel_launch(void* const* d_in, const int* in_sizes, int n_in,
                              void* d_out, int out_size, void* d_ws, size_t ws_size,
                              hipStream_t stream) {
  (void)n_in; (void)out_size; (void)ws_size;

  const float* item   = (const float*)d_in[0];
  const float* action = (const float*)d_in[1];
  const float* c0     = (const float*)d_in[2];
  const float* c1     = (const float*)d_in[3];
  const int*   lens   = (const int*)d_in[4];

  const int Bn      = in_sizes[4];        // batch size (128)
  const int n_items = in_sizes[0] / DD;   // total tokens (131072)
  const int rows    = 2 * n_items;        // interleaved seq rows

  float* out = (float*)d_out;
  int*   off = (int*)d_ws;                // (Bn+1) ints of scratch

  const size_t val_count = (size_t)(rows + 2 * Bn) * DD;
  int* out_len = (int*)d_out + val_count; // second tuple output, int32

  scan_lengths_kernel<<<1, 256, 0, stream>>>(lens, Bn, off, out_len);
  ctx_copy_kernel<<<Bn, DD, 0, stream>>>(c0, c1, off, out);
  const int grid = (rows + 31) / 32;      // 32 rows per block (8 waves x 4)
  seq_copy_kernel<<<grid, 256, 0, stream>>>(item, action, off, out, Bn, rows);
}